// QLSTM_65481071395228
// MI455X (gfx1250) — compile-verified
//
#include <hip/hip_runtime.h>
#include <hip/hip_bf16.h>

#define T_STEPS 256
#define BATCH   256
#define DDIM    128
#define HQ      10
#define FANIN   138   // D + H
#define NPAD    48    // 4*HQ padded to 3 WMMA N-tiles

typedef __attribute__((ext_vector_type(2))) float v2f;
typedef __attribute__((ext_vector_type(8))) float v8f;

// ----------------- wave32 cross-lane helpers -----------------
__device__ __forceinline__ float lshfl(float v, int src) { return __shfl(v, src, 32); }

// xor-shuffle on the VALU path wherever the ISA allows:
//   xor 1/2/4 -> dpp8 permute within 8 lanes
//   xor 8     -> dpp16 row_ror:8 ((l+8)%16 == l^8, involution)
//   xor 16    -> v_permlanex16 identity crossover (lane i <- other half, lane i)
template<int MASK>
__device__ __forceinline__ float xshflx(float v) {
  if constexpr (MASK == 1 || MASK == 2 || MASK == 4) {
    constexpr int sel = ((0 ^ MASK) << 0)  | ((1 ^ MASK) << 3)  | ((2 ^ MASK) << 6)  |
                        ((3 ^ MASK) << 9)  | ((4 ^ MASK) << 12) | ((5 ^ MASK) << 15) |
                        ((6 ^ MASK) << 18) | ((7 ^ MASK) << 21);
    return __int_as_float(__builtin_amdgcn_mov_dpp8(__float_as_int(v), sel));
  } else if constexpr (MASK == 8) {
    return __int_as_float(
        __builtin_amdgcn_update_dpp(0, __float_as_int(v), 0x128, 0xF, 0xF, true));
  } else {
#if __has_builtin(__builtin_amdgcn_permlanex16)
    const int x = __float_as_int(v);
    return __int_as_float(
        __builtin_amdgcn_permlanex16(x, x, 0x76543210, 0xFEDCBA98, true, false));
#else
    return __shfl_xor(v, MASK, 32);
#endif
  }
}

__device__ __forceinline__ float wave_reduce_add(float v) {
  v += xshflx<16>(v);
  v += xshflx<8>(v);
  v += xshflx<4>(v);
  v += xshflx<2>(v);
  v += xshflx<1>(v);
  return v;
}

// RX on a register-bit wire (state bit 5+RB): local 2x2 rotation on register pairs.
// new0 = c*z0 - i*s*z1 ; new1 = -i*s*z0 + c*z1
template<int RB>
__device__ __forceinline__ void rx_reg(float re[32], float im[32], float c, float s) {
  #pragma unroll
  for (int r = 0; r < 32; ++r) {
    if ((r & (1 << RB)) == 0) {
      const int r2 = r | (1 << RB);
      float x0 = re[r], y0 = im[r], x1 = re[r2], y1 = im[r2];
      re[r]  = c * x0 + s * y1;
      im[r]  = c * y0 - s * x1;
      re[r2] = c * x1 + s * y0;
      im[r2] = c * y1 - s * x0;
    }
  }
}

// RX on a lane-bit wire (state bit log2(MASK)): symmetric across the lane pair:
// re' = c*re + s*partner_im ; im' = c*im - s*partner_re
template<int MASK>
__device__ __forceinline__ void rx_lane(float re[32], float im[32], float c, float s) {
  #pragma unroll
  for (int r = 0; r < 32; ++r) {
    float pr = xshflx<MASK>(re[r]);
    float pi = xshflx<MASK>(im[r]);
    float x = re[r], y = im[r];
    re[r] = c * x + s * pi;
    im[r] = c * y - s * pr;
  }
}

// Apply RX on wires 0..9 in order; angle data lives in lane w (cMy/sMy = cos/sin(theta_w/2)).
__device__ __forceinline__ void rx_full_layer(float re[32], float im[32], float cMy, float sMy) {
  float c, s;
  c = lshfl(cMy, 0); s = lshfl(sMy, 0); rx_reg<4>(re, im, c, s);
  c = lshfl(cMy, 1); s = lshfl(sMy, 1); rx_reg<3>(re, im, c, s);
  c = lshfl(cMy, 2); s = lshfl(sMy, 2); rx_reg<2>(re, im, c, s);
  c = lshfl(cMy, 3); s = lshfl(sMy, 3); rx_reg<1>(re, im, c, s);
  c = lshfl(cMy, 4); s = lshfl(sMy, 4); rx_reg<0>(re, im, c, s);
  c = lshfl(cMy, 5); s = lshfl(sMy, 5); rx_lane<16>(re, im, c, s);
  c = lshfl(cMy, 6); s = lshfl(sMy, 6); rx_lane<8>(re, im, c, s);
  c = lshfl(cMy, 7); s = lshfl(sMy, 7); rx_lane<4>(re, im, c, s);
  c = lshfl(cMy, 8); s = lshfl(sMy, 8); rx_lane<2>(re, im, c, s);
  c = lshfl(cMy, 9); s = lshfl(sMy, 9); rx_lane<1>(re, im, c, s);
}

// CNOT chain (0,1)(1,2)...(8,9)(9,0). Wire w <-> state bit (9-w); bits 5..9 are
// register bits 0..4, bits 0..4 are lane bits.
// (0,1)..(3,4): static register swaps.
// (4,5)+(5,6)+(6,7)+(7,8)+(8,9): successive lane permutations fused into ONE
//   gather: src = f56(f67(f78(f89(lane)))), xor 16 extra on odd registers for (4,5).
// (9,0): lane-parity conditional register swap (v_cndmask, no cross-lane traffic).
__device__ __forceinline__ void cnot_chain(float re[32], float im[32], int lane) {
  float t;
  // (0,1): ctrl reg bit4, flip reg bit3 -> swap r <-> r+8 for r in [16,24)
  #pragma unroll
  for (int r = 16; r < 24; ++r) {
    t = re[r]; re[r] = re[r+8]; re[r+8] = t;
    t = im[r]; im[r] = im[r+8]; im[r+8] = t;
  }
  // (1,2): ctrl reg bit3, flip reg bit2
  #pragma unroll
  for (int r = 0; r < 32; ++r) {
    if ((r & 8) && !(r & 4)) {
      t = re[r]; re[r] = re[r+4]; re[r+4] = t;
      t = im[r]; im[r] = im[r+4]; im[r+4] = t;
    }
  }
  // (2,3): ctrl reg bit2, flip reg bit1
  #pragma unroll
  for (int r = 0; r < 32; ++r) {
    if ((r & 4) && !(r & 2)) {
      t = re[r]; re[r] = re[r+2]; re[r+2] = t;
      t = im[r]; im[r] = im[r+2]; im[r+2] = t;
    }
  }
  // (3,4): ctrl reg bit1, flip reg bit0
  #pragma unroll
  for (int r = 0; r < 32; ++r) {
    if ((r & 2) && !(r & 1)) {
      t = re[r]; re[r] = re[r+1]; re[r+1] = t;
      t = im[r]; im[r] = im[r+1]; im[r+1] = t;
    }
  }
  // fused lane-CNOTs: compose f89 -> f78 -> f67 -> f56 (innermost applied first)
  int m = lane ^ ((lane & 2) ? 1 : 0);   // f89
  m ^= (m & 4) ? 2 : 0;                  // f78
  m ^= (m & 8) ? 4 : 0;                  // f67
  m ^= (m & 16) ? 8 : 0;                 // f56
  const int srcE = m;                    // even registers
  const int srcO = m ^ 16;               // odd registers: CNOT(4,5) lane-xor16 folded in
  #pragma unroll
  for (int r = 0; r < 32; ++r) {
    const int s = (r & 1) ? srcO : srcE;
    re[r] = lshfl(re[r], s);
    im[r] = lshfl(im[r], s);
  }
  // (9,0): ctrl lane bit0, flip reg bit4 -> conditional register swap
  {
    bool sel = (lane & 1) != 0;
    #pragma unroll
    for (int r = 0; r < 16; ++r) {
      float a = re[r], b = re[r+16];
      re[r] = sel ? b : a; re[r+16] = sel ? a : b;
      a = im[r]; b = im[r+16];
      im[r] = sel ? b : a; im[r+16] = sel ? a : b;
    }
  }
}

__device__ __forceinline__ float sigmoidf(float x) { return 1.0f / (1.0f + expf(-x)); }

// ----------------- weight packing for WMMA B-panel -----------------
// Bpack[k][n] (n = g*10+w, padded to 48): column n = row w of gate-g weight, x-part (k<128).
__global__ void qlstm_pack(const float* __restrict__ Wf, const float* __restrict__ bf,
                           const float* __restrict__ Wi, const float* __restrict__ bi,
                           const float* __restrict__ Wg, const float* __restrict__ bg,
                           const float* __restrict__ Wo, const float* __restrict__ bo,
                           float* __restrict__ Bpack, float* __restrict__ biasPack) {
  int idx = blockIdx.x * blockDim.x + threadIdx.x;
  if (idx < DDIM * NPAD) {
    int k = idx / NPAD, n = idx % NPAD;
    float v = 0.0f;
    if (n < 4 * HQ) {
      int g = n / HQ, w = n % HQ;
      const float* W = (g == 0) ? Wf : (g == 1) ? Wi : (g == 2) ? Wg : Wo;
      v = W[w * FANIN + k];
    }
    Bpack[idx] = v;
  }
  if (idx < NPAD) {
    float v = 0.0f;
    if (idx < 4 * HQ) {
      int g = idx / HQ, w = idx % HQ;
      const float* bb = (g == 0) ? bf : (g == 1) ? bi : (g == 2) ? bg : bo;
      v = bb[w];
    }
    biasPack[idx] = v;
  }
}

// ----------------- fp32 WMMA pre-activation GEMM -----------------
// A_pre[row][n] = sum_{k<128} X[row][k] * Bpack[k][n] + bias[n], rows = T*B = 65536.
// One wave per 16x16 tile; 32 k-steps of V_WMMA_F32_16X16X4_F32.
__global__ __launch_bounds__(256) void qlstm_pregemm(const float* __restrict__ X,
                                                     const float* __restrict__ Bpack,
                                                     const float* __restrict__ biasPack,
                                                     float* __restrict__ A_pre) {
  const int tid   = threadIdx.x;
  const int lane  = tid & 31;
  const int wave  = blockIdx.x * (blockDim.x >> 5) + (tid >> 5); // 0..12287
  const int mtile = wave / 3;
  const int ntile = wave % 3;
  const int klane = lane >> 4;   // 0: K pair {0,1}, 1: K pair {2,3}
  const int ml    = lane & 15;
  const int row   = mtile * 16 + ml;       // A-matrix row M = lane&15
  const int ncol  = ntile * 16 + ml;       // B/D column N = lane&15

  const float* xrow = X + (size_t)row * DDIM;
  v8f acc = {};
  #pragma unroll 4
  for (int k0 = 0; k0 < DDIM; k0 += 4) {
    const int ka = k0 + klane * 2;
    v2f a, b;
    a.x = xrow[ka];
    a.y = xrow[ka + 1];
    b.x = Bpack[(size_t)ka * NPAD + ncol];
    b.y = Bpack[(size_t)(ka + 1) * NPAD + ncol];
    acc = __builtin_amdgcn_wmma_f32_16x16x4_f32(false, a, false, b, (short)0, acc,
                                                false, false);
  }
  if (ncol < 4 * HQ) {
    const float bias = biasPack[ncol];
    #pragma unroll
    for (int j = 0; j < 8; ++j) {
      const int m = j + klane * 8;          // D: vgpr j = rows j / j+8
      A_pre[(size_t)(mtile * 16 + m) * (4 * HQ) + ncol] = acc[j] + bias;
    }
  }
}

// ----------------- one recurrent time step -----------------
// grid: 256 blocks (batch), block: 128 threads = 4 waves (one per LSTM gate).
// Each wave simulates the full 10-qubit circuit with the state in registers:
// 1024 amplitudes = 32 complex per lane; lane = low 5 index bits, register = high 5.
__global__ __launch_bounds__(128) void qlstm_step(
    const float* __restrict__ inputs, const float* __restrict__ rx_params,
    const float* __restrict__ Wf, const float* __restrict__ bf,
    const float* __restrict__ Wi, const float* __restrict__ bi,
    const float* __restrict__ Wg, const float* __restrict__ bg,
    const float* __restrict__ Wo, const float* __restrict__ bo,
    const float* __restrict__ A_pre,  // [T*B, 40] with bias, or nullptr
    float* __restrict__ hbuf, float* __restrict__ cbuf,
    float* __restrict__ out, int t, int is_last) {
  const int tid  = threadIdx.x;
  const int lane = tid & 31;
  const int g    = tid >> 5;   // gate: 0=f 1=i 2=g 3=o
  const int b    = blockIdx.x;

  const float* W    = (g == 0) ? Wf : (g == 1) ? Wi : (g == 2) ? Wg : Wo;
  const float* bias = (g == 0) ? bf : (g == 1) ? bi : (g == 2) ? bg : bo;

  // ---- angles: A_pre (or x-part) + h @ Wh.T, result in lane w (w<10) ----
  float myang = 0.0f;
  if (A_pre) {
    if (lane < HQ) myang = A_pre[(size_t)(t * BATCH + b) * (4 * HQ) + g * HQ + lane];
  } else {
    const float* xrow = inputs + (size_t)(t * BATCH + b) * DDIM;
    #pragma unroll
    for (int w = 0; w < HQ; ++w) {
      float acc = 0.0f;
      #pragma unroll
      for (int k = 0; k < 4; ++k) {
        const int d = lane + 32 * k;
        acc += xrow[d] * W[w * FANIN + d];
      }
      acc = wave_reduce_add(acc);
      if (lane == w) myang = acc;
    }
    if (lane < HQ) myang += bias[lane];
  }
  if (lane < HQ) {
    const float* hb = hbuf + b * HQ;
    float acc = 0.0f;
    #pragma unroll
    for (int j = 0; j < HQ; ++j) acc += hb[j] * W[lane * FANIN + DDIM + j];
    myang += acc;
  }
  const float cA = cosf(0.5f * myang), sA = sinf(0.5f * myang);
  const float phi = (lane < HQ) ? rx_params[lane] : 0.0f;
  const float cP = cosf(0.5f * phi), sP = sinf(0.5f * phi);

  // ---- state |0...0> ----
  float re[32], im[32];
  #pragma unroll
  for (int r = 0; r < 32; ++r) { re[r] = 0.0f; im[r] = 0.0f; }
  if (lane == 0) re[0] = 1.0f;

  rx_full_layer(re, im, cA, sA);            // data-encoding RX layer
  #pragma unroll
  for (int d = 0; d < 2; ++d) {             // DEPTH variational layers
    rx_full_layer(re, im, cP, sP);
    cnot_chain(re, im, lane);
  }

  // ---- PauliZ expectations: wires 0..4 via register-bit signs, 5..9 via lane-bit signs ----
  float vals[HQ];
  float tot = 0.0f;
  vals[0] = vals[1] = vals[2] = vals[3] = vals[4] = 0.0f;
  #pragma unroll
  for (int r = 0; r < 32; ++r) {
    const float p = re[r] * re[r] + im[r] * im[r];
    tot += p;
    vals[0] += (r & 16) ? -p : p;
    vals[1] += (r & 8)  ? -p : p;
    vals[2] += (r & 4)  ? -p : p;
    vals[3] += (r & 2)  ? -p : p;
    vals[4] += (r & 1)  ? -p : p;
  }
  vals[5] = (lane & 16) ? -tot : tot;
  vals[6] = (lane & 8)  ? -tot : tot;
  vals[7] = (lane & 4)  ? -tot : tot;
  vals[8] = (lane & 2)  ? -tot : tot;
  vals[9] = (lane & 1)  ? -tot : tot;
  #pragma unroll
  for (int k = 0; k < HQ; ++k) vals[k] = wave_reduce_add(vals[k]);

  // ---- combine the 4 gates -> LSTM cell update ----
  __shared__ float gateE[4][HQ];
  if (lane == 0) {
    #pragma unroll
    for (int k = 0; k < HQ; ++k) gateE[g][k] = vals[k];
  }
  __syncthreads();
  if (tid < HQ) {
    const int w = tid;
    const float f  = sigmoidf(gateE[0][w]);
    const float ii = sigmoidf(gateE[1][w]);
    const float gg = tanhf(gateE[2][w]);
    const float o  = sigmoidf(gateE[3][w]);
    const float cp = cbuf[b * HQ + w];
    const float cn = f * cp + ii * gg;
    const float hn = o * tanhf(cn);
    cbuf[b * HQ + w] = cn;
    hbuf[b * HQ + w] = hn;
    out[(size_t)t * BATCH * HQ + b * HQ + w] = hn;
    if (is_last) {
      out[(size_t)T_STEPS * BATCH * HQ + b * HQ + w] = hn;                 // hT
      out[(size_t)T_STEPS * BATCH * HQ + BATCH * HQ + b * HQ + w] = cn;    // cT
    }
  }
}

extern "C" void kernel_launch(void* const* d_in, const int* in_sizes, int n_in,
                              void* d_out, int out_size, void* d_ws, size_t ws_size,
                              hipStream_t stream) {
  const float* inputs = (const float*)d_in[0];
  const float* rx     = (const float*)d_in[1];
  const float* Wf = (const float*)d_in[2]; const float* bf = (const float*)d_in[3];
  const float* Wi = (const float*)d_in[4]; const float* bi = (const float*)d_in[5];
  const float* Wg = (const float*)d_in[6]; const float* bg = (const float*)d_in[7];
  const float* Wo = (const float*)d_in[8]; const float* bo = (const float*)d_in[9];
  float* out = (float*)d_out;

  float* ws       = (float*)d_ws;
  float* hbuf     = ws;                        // 2560 f
  float* cbuf     = ws + BATCH * HQ;           // 2560 f
  float* biasPack = ws + 2 * BATCH * HQ;       // 48 f (pad to 64)
  float* Bpack    = biasPack + 64;             // 128*48 = 6144 f
  float* A_pre    = Bpack + DDIM * NPAD;       // 65536*40 f
  const size_t needA = (size_t)((2 * BATCH * HQ) + 64 + DDIM * NPAD +
                                T_STEPS * BATCH * 4 * HQ) * sizeof(float);
  const bool useA = (ws_size >= needA);

  // zero h0 / c0
  hipMemsetAsync(ws, 0, (size_t)2 * BATCH * HQ * sizeof(float), stream);

  if (useA) {
    qlstm_pack<<<(DDIM * NPAD + 255) / 256, 256, 0, stream>>>(Wf, bf, Wi, bi, Wg, bg,
                                                              Wo, bo, Bpack, biasPack);
    // 4096 M-tiles * 3 N-tiles = 12288 waves; 8 waves/block -> 1536 blocks
    qlstm_pregemm<<<1536, 256, 0, stream>>>(inputs, Bpack, biasPack, A_pre);
  }

  for (int t = 0; t < T_STEPS; ++t) {
    qlstm_step<<<BATCH, 128, 0, stream>>>(inputs, rx, Wf, bf, Wi, bi, Wg, bg, Wo, bo,
                                          useA ? A_pre : nullptr, hbuf, cbuf, out, t,
                                          (t == T_STEPS - 1) ? 1 : 0);
  }
}